// GraphSAGE_84121229460234
// MI455X (gfx1250) — compile-verified
//
#include <hip/hip_runtime.h>
#include <hip/hip_bf16.h>

typedef __attribute__((ext_vector_type(2))) float v2f;
typedef __attribute__((ext_vector_type(8))) float v8f;

// ---------------------------------------------------------------------------
// Zero helpers
// ---------------------------------------------------------------------------
__global__ void zero_kernel(float* __restrict__ p, long long n) {
    long long i = (long long)blockIdx.x * blockDim.x + threadIdx.x;
    if (i < n) p[i] = 0.0f;
}

__global__ void zero4_kernel(float4* __restrict__ p, long long n4) {
    long long i = (long long)blockIdx.x * blockDim.x + threadIdx.x;
    if (i < n4) p[i] = make_float4(0.f, 0.f, 0.f, 0.f);
}

// ---------------------------------------------------------------------------
// Degree: deg[dst[e]] += 1  (float atomics; counts exact below 2^24)
// ---------------------------------------------------------------------------
__global__ void degree_kernel(const int* __restrict__ dst, float* __restrict__ deg,
                              long long E) {
    long long e = (long long)blockIdx.x * blockDim.x + threadIdx.x;
    if (e < E) atomicAdd(&deg[dst[e]], 1.0f);
}

// invdeg[i] = 1 / max(deg[i], 1)
__global__ void invdeg_kernel(float* __restrict__ deg, int N) {
    int i = blockIdx.x * blockDim.x + threadIdx.x;
    if (i < N) deg[i] = 1.0f / fmaxf(deg[i], 1.0f);
}

// ---------------------------------------------------------------------------
// Edge scatter, x4 vectorized: one thread owns 4 consecutive features of one
// edge message: b128 load + 4x global_atomic_add_f32 (resolved in L2).
//   qshift = log2(D/4), shift = log2(D)
// ---------------------------------------------------------------------------
__global__ void scatter_add_kernel(const float* __restrict__ x,
                                   const int* __restrict__ src,
                                   const int* __restrict__ dst,
                                   float* __restrict__ agg,
                                   long long nquads, int qshift, int qmask, int shift) {
    long long idx = (long long)blockIdx.x * blockDim.x + threadIdx.x;
    if (idx >= nquads) return;
    long long e = idx >> qshift;
    int q = (int)(idx & qmask);
    int s = src[e];
    int d = dst[e];
    const float4 v = *(const float4*)(x + ((size_t)s << shift) + 4 * q);
    float* a = agg + ((size_t)d << shift) + 4 * q;
    atomicAdd(a + 0, v.x);
    atomicAdd(a + 1, v.y);
    atomicAdd(a + 2, v.z);
    atomicAdd(a + 3, v.w);
}

// ---------------------------------------------------------------------------
// Fused SAGE GEMM, wave32. One wave computes a full 16-row x NO-col strip:
//   out = act( X[N,K] @ Wself  +  (AGG[N,K] * invdeg[row]) @ Wneigh  + b )
//
// Weights staged in LDS once per 256-thread block (8 waves = 128 rows/block)
// in a K-PAIR INTERLEAVED layout:
//   sW[p*NO + n] = { W[2p][n], W[2p+1][n] }   (float2)
// so each B operand {W[kk][n], W[kk+1][n]} (kk = k + khalf, always even) is a
// single aligned ds_load_b64 straight into an even VGPR pair — no re-pairing
// movs in the inner loop. Two passes (Wself then Wneigh) reuse one buffer.
//
// V_WMMA_F32_16X16X4_F32 — exact fp32 MMA; NT=NO/16 accumulator tiles/wave.
// VGPR layouts (ISA 7.12.2):
//   A 16x4 f32 : lane holds A[m0 + lane%16][kk + {0,1}]  (contiguous float2)
//   B 4x16 f32 : lane holds B[kk + {0,1}][n0 + lane%16]
//   C/D 16x16  : reg v of lane = D[m0 + v + 8*(lane/16)][n0 + lane%16]
// No early returns: EXEC is all-1s at every WMMA; clamped waves duplicate
// another strip's work and write identical values (benign, deterministic).
// ---------------------------------------------------------------------------
template <int K, int NO, bool HAS_NEIGH, bool RELU>
__global__ void __launch_bounds__(256)
sage_gemm_kernel(const float* __restrict__ X, const float* __restrict__ AGG,
                 const float* __restrict__ invdeg,
                 const float* __restrict__ Wself, const float* __restrict__ Wneigh,
                 const float* __restrict__ bias, float* __restrict__ out, int N) {
    constexpr int NT = NO / 16;
    __shared__ v2f   sW[(K / 2) * NO];   // k-pair interleaved, <= 64KB
    __shared__ float sBias[NO];

    const int lane   = threadIdx.x & 31;
    const int wave   = threadIdx.x >> 5;
    const int mtiles = (N + 15) >> 4;
    int gw = blockIdx.x * (blockDim.x >> 5) + wave;
    if (gw >= mtiles) gw = mtiles - 1;           // clamp: no divergent exit
    const int m0    = gw << 4;
    const int l15   = lane & 15;
    const int mrow  = m0 + l15;
    const int row   = (mrow < N) ? mrow : (N - 1);
    const int khalf = (lane >> 4) << 1;          // 0 or 2

    float* sWf = (float*)sW;

    // ---- stage Wself (interleaved) + bias into LDS ----------------------
    for (int i = threadIdx.x; i < K * NO; i += blockDim.x) {
        const int k = i / NO;                    // NO is a power of two
        const int n = i & (NO - 1);
        sWf[(((k >> 1) * NO + n) << 1) | (k & 1)] = Wself[i];
    }
    for (int i = threadIdx.x; i < NO; i += blockDim.x) sBias[i] = bias[i];
    __syncthreads();

    // ---- init accumulators with bias (C = b broadcast over rows) --------
    v8f acc[NT];
#pragma unroll
    for (int t = 0; t < NT; ++t) {
        const float bv = sBias[t * 16 + l15];
#pragma unroll
        for (int v = 0; v < 8; ++v) acc[t][v] = bv;
    }

    // ---- pass 1: X @ Wself ----------------------------------------------
    {
        const float* xrow = X + (size_t)row * K;
        for (int k = 0; k < K; k += 4) {
            v2f a = *(const v2f*)(xrow + k + khalf);
            const v2f* brow = sW + ((k + khalf) >> 1) * NO + l15;
#pragma unroll
            for (int t = 0; t < NT; ++t) {
                v2f b = brow[t * 16];            // single ds_load_b64
                acc[t] = __builtin_amdgcn_wmma_f32_16x16x4_f32(
                    false, a, false, b, (short)0, acc[t], false, false);
            }
        }
    }

    // ---- pass 2: (AGG * invdeg) @ Wneigh --------------------------------
    if constexpr (HAS_NEIGH) {
        __syncthreads();   // pass-1 LDS reads done
        for (int i = threadIdx.x; i < K * NO; i += blockDim.x) {
            const int k = i / NO;
            const int n = i & (NO - 1);
            sWf[(((k >> 1) * NO + n) << 1) | (k & 1)] = Wneigh[i];
        }
        __syncthreads();
        const float inv = invdeg[row];
        const float* arow = AGG + (size_t)row * K;
        for (int k = 0; k < K; k += 4) {
            v2f a = *(const v2f*)(arow + k + khalf);
            a.x *= inv;
            a.y *= inv;
            const v2f* brow = sW + ((k + khalf) >> 1) * NO + l15;
#pragma unroll
            for (int t = 0; t < NT; ++t) {
                v2f b = brow[t * 16];
                acc[t] = __builtin_amdgcn_wmma_f32_16x16x4_f32(
                    false, a, false, b, (short)0, acc[t], false, false);
            }
        }
    }

    // ---- epilogue: optional ReLU, guarded stores ------------------------
    const int mbase = m0 + ((lane >> 4) << 3);
#pragma unroll
    for (int t = 0; t < NT; ++t) {
        const int col = t * 16 + l15;
#pragma unroll
        for (int v = 0; v < 8; ++v) {
            float r = acc[t][v];
            if constexpr (RELU) r = fmaxf(r, 0.0f);
            const int rr = mbase + v;
            if (rr < N) out[(size_t)rr * NO + col] = r;
        }
    }
}

// ---------------------------------------------------------------------------
// Launch
// ---------------------------------------------------------------------------
extern "C" void kernel_launch(void* const* d_in, const int* in_sizes, int n_in,
                              void* d_out, int out_size, void* d_ws, size_t ws_size,
                              hipStream_t stream) {
    (void)n_in; (void)out_size; (void)ws_size;

    const float* features = (const float*)d_in[0];   // [N,64]
    const float* W_neigh1 = (const float*)d_in[1];   // [64,128]
    const float* W_self1  = (const float*)d_in[2];   // [64,128]
    const float* b1       = (const float*)d_in[3];   // [128]
    const float* W_neigh2 = (const float*)d_in[4];   // [128,128]
    const float* W_self2  = (const float*)d_in[5];   // [128,128]
    const float* b2       = (const float*)d_in[6];   // [128]
    const float* Wm1      = (const float*)d_in[7];   // [128,128]
    const float* bm1      = (const float*)d_in[8];   // [128]
    const float* Wm2      = (const float*)d_in[9];   // [128,64]
    const float* bm2      = (const float*)d_in[10];  // [64]
    const int*   src      = (const int*)d_in[11];    // [E]
    const int*   dst      = (const int*)d_in[12];    // [E]

    const int       IN_F = 64, HID = 128;
    const int       N = in_sizes[0] / IN_F;
    const long long E = in_sizes[11];

    // Workspace layout (floats, 256B-aligned chunks)
    size_t Np = ((size_t)N + 63) & ~(size_t)63;
    float* deg = (float*)d_ws;               // N      (deg -> invdeg in place)
    float* agg = deg + Np;                   // N*128  (layer1 uses N*64 of it)
    float* hA  = agg + Np * (size_t)HID;     // N*128
    float* hB  = hA  + Np * (size_t)HID;     // N*128

    const int T = 256;
    auto blocks = [](long long n, int t) { return (unsigned)((n + t - 1) / t); };
    const int mtiles = (N + 15) / 16;
    const unsigned gemm_blk = (unsigned)((mtiles + (T / 32) - 1) / (T / 32));

    // ---- degrees ---------------------------------------------------------
    zero_kernel<<<blocks(N, T), T, 0, stream>>>(deg, N);
    degree_kernel<<<blocks(E, T), T, 0, stream>>>(dst, deg, E);
    invdeg_kernel<<<blocks(N, T), T, 0, stream>>>(deg, N);

    // ---- layer 1: aggregate features (D=64), fused SAGE GEMM ------------
    zero4_kernel<<<blocks((long long)N * IN_F / 4, T), T, 0, stream>>>(
        (float4*)agg, (long long)N * IN_F / 4);
    {
        long long nq = E * (IN_F / 4);
        scatter_add_kernel<<<blocks(nq, T), T, 0, stream>>>(
            features, src, dst, agg, nq, /*qshift=*/4, /*qmask=*/IN_F / 4 - 1, /*shift=*/6);
    }
    sage_gemm_kernel<64, 128, true, true><<<gemm_blk, T, 0, stream>>>(
        features, agg, deg, W_self1, W_neigh1, b1, hA, N);

    // ---- layer 2: aggregate hA (D=128), fused SAGE GEMM -----------------
    zero4_kernel<<<blocks((long long)N * HID / 4, T), T, 0, stream>>>(
        (float4*)agg, (long long)N * HID / 4);
    {
        long long nq = E * (HID / 4);
        scatter_add_kernel<<<blocks(nq, T), T, 0, stream>>>(
            hA, src, dst, agg, nq, /*qshift=*/5, /*qmask=*/HID / 4 - 1, /*shift=*/7);
    }
    sage_gemm_kernel<128, 128, true, true><<<gemm_blk, T, 0, stream>>>(
        hA, agg, deg, W_self2, W_neigh2, b2, hB, N);

    // ---- MLP layer 1: relu(hB @ Wm1 + bm1) -> hA ------------------------
    sage_gemm_kernel<128, 128, false, true><<<gemm_blk, T, 0, stream>>>(
        hB, nullptr, deg, Wm1, nullptr, bm1, hA, N);

    // ---- MLP layer 2: hA @ Wm2 + bm2 -> d_out ---------------------------
    sage_gemm_kernel<128, 64, false, false><<<gemm_blk, T, 0, stream>>>(
        hA, nullptr, deg, Wm2, nullptr, bm2, (float*)d_out, N);
}